// ResidualLoss_12146167513806
// MI455X (gfx1250) — compile-verified
//
#include <hip/hip_runtime.h>
#include <hip/hip_bf16.h>

// CDNA5 / gfx1250. wave32.
typedef float v2f __attribute__((ext_vector_type(2)));
typedef float v8f __attribute__((ext_vector_type(8)));

// Relaxed agent-scope f32 atomic add, no return -> global_atomic_add_f32 (STOREcnt path)
__device__ __forceinline__ void atomicAddF32(float* p, float v) {
    __hip_atomic_fetch_add(p, v, __ATOMIC_RELAXED, __HIP_MEMORY_SCOPE_AGENT);
}

// ---------------------------------------------------------------------------
// Wave32 sum via the matrix pipe: V_WMMA_F32_16X16X4_F32.
// A (16x4 f32, 2 VGPRs/lane) holds {v, 0}; B = all ones; C = 0.
// D[m][n] = rowsum(A)[m] replicated across 16 columns, so
//   sum over ALL D entries (8 VGPRs x 32 lanes) == 16 * sum_lanes(v)
// for ANY bijective (m,k)<->(lane,vgpr) mapping. Then 5x shfl_xor gives the
// all-lane total of the per-lane d-sums; *1/16 (exact) recovers sum(v).
// EXEC must be all ones: callers use full blocks with converged control flow.
// ---------------------------------------------------------------------------
__device__ __forceinline__ float wave_sum_wmma(float v) {
    v2f a; a.x = v;     a.y = 0.0f;
    v2f b; b.x = 1.0f;  b.y = 1.0f;
    v8f c = {};
    v8f d = __builtin_amdgcn_wmma_f32_16x16x4_f32(
        /*neg_a=*/false, a, /*neg_b=*/false, b,
        /*c_mod=*/(short)0, c, /*reuse_a=*/false, /*reuse_b=*/false);
    float s = d[0] + d[1] + d[2] + d[3] + d[4] + d[5] + d[6] + d[7];
    s += __shfl_xor(s, 1, 32);
    s += __shfl_xor(s, 2, 32);
    s += __shfl_xor(s, 4, 32);
    s += __shfl_xor(s, 8, 32);
    s += __shfl_xor(s, 16, 32);
    return s * 0.0625f;   // exact power-of-two scale
}

// ---------------------------------------------------------------------------
// Kernel 0: zero F_int (3N floats) + the two scalar accumulators (contiguous)
// ---------------------------------------------------------------------------
__global__ void k_zero(float* __restrict__ ws, long n) {
    long stride = (long)gridDim.x * blockDim.x;
    for (long i = (long)blockIdx.x * blockDim.x + threadIdx.x; i < n; i += stride)
        ws[i] = 0.0f;
}

// ---------------------------------------------------------------------------
// Kernel 1: per-element beam forces, scatter-add into F_int.
// Streamed element data (~36B/elem) + L2-resident random gathers of pred_raw.
// ---------------------------------------------------------------------------
__global__ void k_elem_forces(const float* __restrict__ pred,
                              const float* __restrict__ u_c,
                              const float* __restrict__ th_c,
                              const int2*  __restrict__ conn,
                              const float* __restrict__ Lp,
                              const float* __restrict__ pE,
                              const float* __restrict__ pA,
                              const float* __restrict__ pI,
                              const float* __restrict__ dirs,
                              float* __restrict__ Fint,
                              int E) {
    const float uc = u_c[0];
    const float tc = th_c[0];
    const int stride = gridDim.x * blockDim.x;
    for (int e = blockIdx.x * blockDim.x + threadIdx.x; e < E; e += stride) {
        // gfx1250 global_prefetch_b8 on next iteration's streamed data
        int en = e + stride;
        if (en < E) {
            __builtin_prefetch(&conn[en],      0, 1);
            __builtin_prefetch(&Lp[en],        0, 1);
            __builtin_prefetch(&dirs[3 * en],  0, 1);
        }

        const int2 ab = conn[e];
        const int nA = ab.x, nB = ab.y;
        const float c = dirs[3 * e + 0];
        const float s = dirs[3 * e + 2];
        const float l  = Lp[e];
        const float Ee = pE[e];
        const float EA = Ee * pA[e];
        const float EI = Ee * pI[e];
        const float invl  = 1.0f / l;
        const float ea_l  = EA * invl;
        const float ei_l  = EI * invl;
        const float ei_l2 = ei_l  * invl;
        const float ei_l3 = ei_l2 * invl;

        const float uA0 = pred[3 * nA + 0] * uc;
        const float uA1 = pred[3 * nA + 1] * uc;
        const float uA2 = pred[3 * nA + 2] * tc;
        const float uB0 = pred[3 * nB + 0] * uc;
        const float uB1 = pred[3 * nB + 1] * uc;
        const float uB2 = pred[3 * nB + 2] * tc;

        const float u_A =  c * uA0 + s * uA1;
        const float w_A = -s * uA0 + c * uA1;
        const float thA = -uA2;
        const float u_B =  c * uB0 + s * uB1;
        const float w_B = -s * uB0 + c * uB1;
        const float thB = -uB2;

        const float dw = w_A - w_B;
        const float f0 = ea_l * (u_A - u_B);                               // AXIAL_WEIGHT = 1
        const float f1 = 12.0f * ei_l3 * dw + 6.0f * ei_l2 * (thA + thB);
        const float f2 =  6.0f * ei_l2 * dw + 4.0f * ei_l * thA + 2.0f * ei_l * thB;
        const float f5 =  6.0f * ei_l2 * dw + 2.0f * ei_l * thA + 4.0f * ei_l * thB;

        // fA = (c*f0 - s*f1, s*f0 + c*f1, -f2)
        atomicAddF32(&Fint[3 * nA + 0],  c * f0 - s * f1);
        atomicAddF32(&Fint[3 * nA + 1],  s * f0 + c * f1);
        atomicAddF32(&Fint[3 * nA + 2], -f2);
        // fB = (c*f3 - s*f4, s*f3 + c*f4, -f5), f3 = -f0, f4 = -f1
        atomicAddF32(&Fint[3 * nB + 0], -(c * f0) + s * f1);
        atomicAddF32(&Fint[3 * nB + 1], -(s * f0) - c * f1);
        atomicAddF32(&Fint[3 * nB + 2], -f5);
    }
}

// ---------------------------------------------------------------------------
// Kernel 2: per-node masked residual; accumulate num = sum(R_free^2) and
// den = sum(F_free^2). Per-wave WMMA reduction, one f32 atomic per wave.
// ---------------------------------------------------------------------------
__global__ void k_node_reduce(const float* __restrict__ Fint,
                              const float* __restrict__ Fext,
                              const float* __restrict__ bcd,
                              const float* __restrict__ bcr,
                              float* __restrict__ acc,  // acc[0]=num, acc[1]=den
                              int N) {
    float num = 0.0f, den = 0.0f;
    const int stride = gridDim.x * blockDim.x;
    for (int n = blockIdx.x * blockDim.x + threadIdx.x; n < N; n += stride) {
        const float md = 1.0f - bcd[n];
        const float mr = 1.0f - bcr[n];
        const float e0 = Fext[3 * n + 0];
        const float e1 = Fext[3 * n + 1];
        const float e2 = Fext[3 * n + 2];
        const float r0 = (Fint[3 * n + 0] - e0) * md;
        const float r1 = (Fint[3 * n + 1] - e1) * md;
        const float r2 = (Fint[3 * n + 2] - e2) * mr;
        const float g0 = e0 * md, g1 = e1 * md, g2 = e2 * mr;
        num += r0 * r0 + r1 * r1 + r2 * r2;
        den += g0 * g0 + g1 * g1 + g2 * g2;
    }
    // all lanes converged here; EXEC is all ones for the WMMA
    num = wave_sum_wmma(num);
    den = wave_sum_wmma(den);
    if ((threadIdx.x & 31) == 0) {
        atomicAddF32(&acc[0], num);
        atomicAddF32(&acc[1], den);
    }
}

// ---------------------------------------------------------------------------
// Kernel 3: loss = num / max(den, 1e-30)
// ---------------------------------------------------------------------------
__global__ void k_finalize(const float* __restrict__ acc, float* __restrict__ out) {
    out[0] = acc[0] / fmaxf(acc[1], 1e-30f);
}

extern "C" void kernel_launch(void* const* d_in, const int* in_sizes, int n_in,
                              void* d_out, int out_size, void* d_ws, size_t ws_size,
                              hipStream_t stream) {
    // Input order (setup_inputs dict):
    // 0 pred_raw(3N) 1 u_c(1) 2 theta_c(1) 3 elem_lengths(E) 4 prop_E(E)
    // 5 prop_A(E) 6 prop_I22(E) 7 elem_directions(3E) 8 F_ext(3N)
    // 9 bc_disp(N) 10 bc_rot(N) 11 connectivity(2E, int32)
    const float* pred = (const float*)d_in[0];
    const float* u_c  = (const float*)d_in[1];
    const float* th_c = (const float*)d_in[2];
    const float* Lp   = (const float*)d_in[3];
    const float* pE   = (const float*)d_in[4];
    const float* pA   = (const float*)d_in[5];
    const float* pI   = (const float*)d_in[6];
    const float* dirs = (const float*)d_in[7];
    const float* Fext = (const float*)d_in[8];
    const float* bcd  = (const float*)d_in[9];
    const float* bcr  = (const float*)d_in[10];
    const int2*  conn = (const int2*)d_in[11];

    const int N = in_sizes[9];      // nodes
    const int E = in_sizes[3];      // elements

    float* Fint = (float*)d_ws;                 // 3N floats
    float* acc  = Fint + (size_t)3 * N;         // 2 floats (num, den)

    const long zn = 3L * N + 2;
    k_zero<<<2048, 256, 0, stream>>>((float*)d_ws, zn);
    k_elem_forces<<<4096, 256, 0, stream>>>(pred, u_c, th_c, conn, Lp, pE, pA, pI,
                                            dirs, Fint, E);
    k_node_reduce<<<2048, 256, 0, stream>>>(Fint, Fext, bcd, bcr, acc, N);
    k_finalize<<<1, 1, 0, stream>>>(acc, (float*)d_out);
}